// GAT_19155554140399
// MI455X (gfx1250) — compile-verified
//
#include <hip/hip_runtime.h>
#include <hip/hip_bf16.h>
#include <math.h>

#define NODES 100000
#define EDGES 3200000
#define ETOT  (EDGES + NODES)     // with self-loops appended
#define INC   128
#define HID   64
#define OUTC  16
#define NEG_SLOPE 0.2f

typedef float v2f __attribute__((ext_vector_type(2)));
typedef float v8f __attribute__((ext_vector_type(8)));

// ---------------------------------------------------------------- utilities

__global__ void fill_f32(float* __restrict__ p, float val, int n) {
  int i = blockIdx.x * blockDim.x + threadIdx.x;
  if (i < n) p[i] = val;
}

__device__ __forceinline__ void atomicMaxF(float* addr, float val) {
  // monotonic float-bits trick; works with -inf init, mixed signs
  if (val >= 0.0f) atomicMax((int*)addr, __float_as_int(val));
  else             atomicMin((unsigned int*)addr, __float_as_uint(val));
}

__device__ __forceinline__ void edge_nodes(const long long* __restrict__ ei,
                                           int e, int& u, int& v) {
  if (e < EDGES) { u = (int)ei[e]; v = (int)ei[(size_t)EDGES + e]; }
  else           { u = v = e - EDGES; }  // appended self-loop
}

__device__ __forceinline__ float lrelu(float x) {
  return x > 0.0f ? x : NEG_SLOPE * x;
}

// ------------------------------------------------- GEMM1: h1 = x @ W1 (f32 WMMA)
// Block = 128 threads = 4 waves; wave w owns 16-column tile w; block owns 16 rows.
__global__ __launch_bounds__(128) void gemm1_wmma(const float* __restrict__ x,
                                                  const float* __restrict__ W,
                                                  float* __restrict__ h) {
  const int lane    = threadIdx.x & 31;
  const int wave    = threadIdx.x >> 5;
  const int rowBase = blockIdx.x * 16;        // N % 16 == 0
  const int colBase = wave * 16;
  const int n     = lane & 15;                // A row / B,C column
  const int khalf = (lane >> 4) * 2;          // K split: lanes 0-15 -> K0,K1; 16-31 -> K2,K3
  v8f acc = {};
  const float* xrow = x + (size_t)(rowBase + n) * INC;
  for (int k0 = 0; k0 < INC; k0 += 4) {
    v2f a, b;
    a.x = xrow[k0 + khalf];
    a.y = xrow[k0 + khalf + 1];
    b.x = W[(size_t)(k0 + khalf) * HID + colBase + n];
    b.y = W[(size_t)(k0 + khalf + 1) * HID + colBase + n];
    acc = __builtin_amdgcn_wmma_f32_16x16x4_f32(false, a, false, b,
                                                (short)0, acc, false, false);
  }
  const int mOff = (lane >> 4) * 8;           // C/D: M = v + 8*(lane>=16)
#pragma unroll
  for (int v = 0; v < 8; ++v)
    h[(size_t)(rowBase + mOff + v) * HID + colBase + n] = acc[v];
}

// ------------------------------------------------- GEMM2: h2 = relu(agg1) @ W2
// 4 node-tiles per block, one wave each; single 16-wide column tile, K = 64.
__global__ __launch_bounds__(128) void gemm2_wmma(const float* __restrict__ hr,
                                                  const float* __restrict__ W,
                                                  float* __restrict__ h2,
                                                  int numTiles) {
  const int lane = threadIdx.x & 31;
  const int wave = threadIdx.x >> 5;
  const int tile = blockIdx.x * 4 + wave;
  if (tile >= numTiles) return;               // wave-uniform branch
  const int rowBase = tile * 16;
  const int n     = lane & 15;
  const int khalf = (lane >> 4) * 2;
  v8f acc = {};
  const float* arow = hr + (size_t)(rowBase + n) * HID;
  for (int k0 = 0; k0 < HID; k0 += 4) {
    v2f a, b;
    a.x = arow[k0 + khalf];
    a.y = arow[k0 + khalf + 1];
    b.x = W[(k0 + khalf) * OUTC + n];
    b.y = W[(k0 + khalf + 1) * OUTC + n];
    acc = __builtin_amdgcn_wmma_f32_16x16x4_f32(false, a, false, b,
                                                (short)0, acc, false, false);
  }
  const int mOff = (lane >> 4) * 8;
#pragma unroll
  for (int v = 0; v < 8; ++v)
    h2[(size_t)(rowBase + mOff + v) * OUTC + n] = acc[v];
}

// ------------------------------- per-node attention halves: s = h.a_src, d = h.a_dst
__global__ __launch_bounds__(256) void node_sd64(const float* __restrict__ h,
                                                 const float* __restrict__ asrc,
                                                 const float* __restrict__ adst,
                                                 float* __restrict__ s,
                                                 float* __restrict__ d) {
  int waveId = (blockIdx.x * blockDim.x + threadIdx.x) >> 5;
  int lane   = threadIdx.x & 31;
  if (waveId >= NODES) return;
  const float* row = h + (size_t)waveId * HID;
  float v0 = row[lane], v1 = row[lane + 32];
  float ps = v0 * asrc[lane] + v1 * asrc[lane + 32];
  float pd = v0 * adst[lane] + v1 * adst[lane + 32];
#pragma unroll
  for (int off = 16; off > 0; off >>= 1) {
    ps += __shfl_xor(ps, off, 32);
    pd += __shfl_xor(pd, off, 32);
  }
  if (lane == 0) { s[waveId] = ps; d[waveId] = pd; }
}

__global__ __launch_bounds__(256) void node_sd16(const float* __restrict__ h,
                                                 const float* __restrict__ asrc,
                                                 const float* __restrict__ adst,
                                                 float* __restrict__ s,
                                                 float* __restrict__ d) {
  int n = blockIdx.x * blockDim.x + threadIdx.x;
  if (n >= NODES) return;
  const float* row = h + (size_t)n * OUTC;
  float ps = 0.f, pd = 0.f;
#pragma unroll
  for (int c = 0; c < OUTC; ++c) { float v = row[c]; ps += v * asrc[c]; pd += v * adst[c]; }
  s[n] = ps; d[n] = pd;
}

// ------------------------------- edge passes: segment max / exp-sum
__global__ __launch_bounds__(256) void edge_max(const long long* __restrict__ ei,
                                                const float* __restrict__ s,
                                                const float* __restrict__ d,
                                                float* __restrict__ m) {
  int e = blockIdx.x * blockDim.x + threadIdx.x;
  if (e >= ETOT) return;
  int u, v; edge_nodes(ei, e, u, v);
  atomicMaxF(&m[v], lrelu(s[u] + d[v]));
}

__global__ __launch_bounds__(256) void fix_m(float* __restrict__ m) {
  int n = blockIdx.x * blockDim.x + threadIdx.x;
  if (n >= NODES) return;
  float v = m[n];
  if (!(v > -1e38f)) m[n] = 0.0f;   // where(isfinite(m), m, 0)
}

__global__ __launch_bounds__(256) void edge_sum(const long long* __restrict__ ei,
                                                const float* __restrict__ s,
                                                const float* __restrict__ d,
                                                const float* __restrict__ m,
                                                float* __restrict__ z) {
  int e = blockIdx.x * blockDim.x + threadIdx.x;
  if (e >= ETOT) return;
  int u, v; edge_nodes(ei, e, u, v);
  atomicAdd(&z[v], expf(lrelu(s[u] + d[v]) - m[v]));
}

// ------------------------------- weighted scatter-aggregate (wave per edge, 64 ch)
__global__ __launch_bounds__(256) void edge_agg64(const long long* __restrict__ ei,
                                                  const float* __restrict__ s,
                                                  const float* __restrict__ d,
                                                  const float* __restrict__ m,
                                                  const float* __restrict__ z,
                                                  const float* __restrict__ h,
                                                  float* __restrict__ agg) {
  int e    = (blockIdx.x * blockDim.x + threadIdx.x) >> 5;
  int lane = threadIdx.x & 31;
  if (e >= ETOT) return;
  int u, v; edge_nodes(ei, e, u, v);
  float coef = expf(lrelu(s[u] + d[v]) - m[v]) / (z[v] + 1e-16f);
  const float* hrow = h + (size_t)u * HID;
  float*       arow = agg + (size_t)v * HID;
  atomicAdd(&arow[lane],      coef * hrow[lane]);
  atomicAdd(&arow[lane + 32], coef * hrow[lane + 32]);
}

// half-wave per edge, 16 channels
__global__ __launch_bounds__(256) void edge_agg16(const long long* __restrict__ ei,
                                                  const float* __restrict__ s,
                                                  const float* __restrict__ d,
                                                  const float* __restrict__ m,
                                                  const float* __restrict__ z,
                                                  const float* __restrict__ h,
                                                  float* __restrict__ agg) {
  int e    = (blockIdx.x * blockDim.x + threadIdx.x) >> 4;
  int lane = threadIdx.x & 15;
  if (e >= ETOT) return;
  int u, v; edge_nodes(ei, e, u, v);
  float coef = expf(lrelu(s[u] + d[v]) - m[v]) / (z[v] + 1e-16f);
  atomicAdd(&agg[(size_t)v * OUTC + lane], coef * h[(size_t)u * OUTC + lane]);
}

// ------------------------------- epilogues
__global__ __launch_bounds__(256) void relu_bias64(float* __restrict__ a,
                                                   const float* __restrict__ b) {
  int i = blockIdx.x * blockDim.x + threadIdx.x;
  if (i >= NODES * HID) return;
  a[i] = fmaxf(a[i] + b[i & (HID - 1)], 0.0f);
}

__global__ __launch_bounds__(256) void log_softmax16(const float* __restrict__ agg,
                                                     const float* __restrict__ b,
                                                     float* __restrict__ out) {
  int n = blockIdx.x * blockDim.x + threadIdx.x;
  if (n >= NODES) return;
  float v[OUTC];
  float mx = -INFINITY;
#pragma unroll
  for (int c = 0; c < OUTC; ++c) { v[c] = agg[(size_t)n * OUTC + c] + b[c]; mx = fmaxf(mx, v[c]); }
  float sum = 0.0f;
#pragma unroll
  for (int c = 0; c < OUTC; ++c) sum += expf(v[c] - mx);
  float ls = logf(sum) + mx;
#pragma unroll
  for (int c = 0; c < OUTC; ++c) out[(size_t)n * OUTC + c] = v[c] - ls;
}

// ---------------------------------------------------------------- launch

extern "C" void kernel_launch(void* const* d_in, const int* in_sizes, int n_in,
                              void* d_out, int out_size, void* d_ws, size_t ws_size,
                              hipStream_t stream) {
  const float*     x     = (const float*)d_in[0];
  const long long* ei    = (const long long*)d_in[1];   // int64 [2, E]
  const float*     W1    = (const float*)d_in[2];
  const float*     as1   = (const float*)d_in[3];
  const float*     ad1   = (const float*)d_in[4];
  const float*     b1    = (const float*)d_in[5];
  const float*     W2    = (const float*)d_in[6];
  const float*     as2   = (const float*)d_in[7];
  const float*     ad2   = (const float*)d_in[8];
  const float*     b2    = (const float*)d_in[9];
  float*           out   = (float*)d_out;

  // workspace layout (floats)
  float* ws   = (float*)d_ws;
  size_t off  = 0;
  float* h1   = ws + off; off += (size_t)NODES * HID;   // 6.4M
  float* s1   = ws + off; off += NODES;
  float* d1   = ws + off; off += NODES;
  float* m1   = ws + off; off += NODES;
  float* z1   = ws + off; off += NODES;
  float* agg1 = ws + off; off += (size_t)NODES * HID;   // relu'd in place
  float* h2   = ws + off; off += (size_t)NODES * OUTC;
  float* s2   = ws + off; off += NODES;
  float* d2   = ws + off; off += NODES;
  float* m2   = ws + off; off += NODES;
  float* z2   = ws + off; off += NODES;
  float* agg2 = ws + off; off += (size_t)NODES * OUTC;
  if (ws_size < off * sizeof(float)) return;

  const int B = 256;
  const int gN     = (NODES + B - 1) / B;
  const int gE     = (ETOT + B - 1) / B;
  const int gNH    = (NODES * HID + B - 1) / B;
  const int gNO    = (NODES * OUTC + B - 1) / B;
  const int gEw32  = ((size_t)ETOT * 32 + B - 1) / B;   // wave per edge
  const int gEw16  = ((size_t)ETOT * 16 + B - 1) / B;   // half-wave per edge
  const int tiles  = NODES / 16;                        // 6250

  // ---- init scratch
  fill_f32<<<gNH, B, 0, stream>>>(agg1, 0.0f, NODES * HID);
  fill_f32<<<gNO, B, 0, stream>>>(agg2, 0.0f, NODES * OUTC);
  fill_f32<<<gN,  B, 0, stream>>>(z1, 0.0f, NODES);
  fill_f32<<<gN,  B, 0, stream>>>(z2, 0.0f, NODES);
  fill_f32<<<gN,  B, 0, stream>>>(m1, -INFINITY, NODES);
  fill_f32<<<gN,  B, 0, stream>>>(m2, -INFINITY, NODES);

  // ---- layer 1
  gemm1_wmma<<<tiles, 128, 0, stream>>>(x, W1, h1);
  node_sd64<<<(NODES * 32 + B - 1) / B, B, 0, stream>>>(h1, as1, ad1, s1, d1);
  edge_max<<<gE, B, 0, stream>>>(ei, s1, d1, m1);
  fix_m<<<gN, B, 0, stream>>>(m1);
  edge_sum<<<gE, B, 0, stream>>>(ei, s1, d1, m1, z1);
  edge_agg64<<<gEw32, B, 0, stream>>>(ei, s1, d1, m1, z1, h1, agg1);
  relu_bias64<<<gNH, B, 0, stream>>>(agg1, b1);

  // ---- layer 2
  gemm2_wmma<<<(tiles + 3) / 4, 128, 0, stream>>>(agg1, W2, h2, tiles);
  node_sd16<<<gN, B, 0, stream>>>(h2, as2, ad2, s2, d2);
  edge_max<<<gE, B, 0, stream>>>(ei, s2, d2, m2);
  fix_m<<<gN, B, 0, stream>>>(m2);
  edge_sum<<<gE, B, 0, stream>>>(ei, s2, d2, m2, z2);
  edge_agg16<<<gEw16, B, 0, stream>>>(ei, s2, d2, m2, z2, h2, agg2);

  // ---- log_softmax -> d_out
  log_softmax16<<<gN, B, 0, stream>>>(agg2, b2, out);
}